// ChebnetClassifier_SHREC14_61357902791050
// MI455X (gfx1250) — compile-verified
//
#include <hip/hip_runtime.h>
#include <hip/hip_bf16.h>
#include <math.h>

// Problem sizes (fixed by the reference)
#define CN0 100000
#define CN1 25000
#define CN2 6250
#define CE0 1600000
#define CE1 400000
#define CE2 100000
#define KCH 6
#define NCLS 40

typedef __attribute__((ext_vector_type(2))) float v2f;
typedef __attribute__((ext_vector_type(8))) float v8f;

// ---------------------------------------------------------------------------
// Elementwise / sparse helpers
// ---------------------------------------------------------------------------

__global__ void fill_f32(float* __restrict__ p, float v, int n) {
  int i = blockIdx.x * blockDim.x + threadIdx.x;
  if (i < n) p[i] = v;
}

__global__ void deg_accum(const int* __restrict__ rows, float* __restrict__ deg, int E) {
  int e = blockIdx.x * blockDim.x + threadIdx.x;
  if (e < E) atomicAdd(&deg[rows[e]], 1.0f);
}

__global__ void deg_to_dinv(float* __restrict__ d, int n) {
  int i = blockIdx.x * blockDim.x + threadIdx.x;
  if (i < n) {
    float deg = d[i];
    d[i] = (deg > 0.0f) ? rsqrtf(fmaxf(deg, 1e-12f)) : 0.0f;
  }
}

// dst[r*ldD + f] = src[r*ldS + f]
__global__ void copy_rows(const float* __restrict__ src, int ldS,
                          float* __restrict__ dst, int ldD, int n, int F) {
  int i = blockIdx.x * blockDim.x + threadIdx.x;
  if (i < n * F) {
    int r = i / F, f = i - r * F;
    dst[r * ldD + f] = src[r * ldS + f];
  }
}

// Tx[:, dstOff..] += norm(e) * Tx[:, srcOff..] scattered over edges.
// norm(e) = -dinv[row]*dinv[col]  (L_hat = -D^-1/2 A D^-1/2)
__global__ void lhat_scatter(const int* __restrict__ rows, const int* __restrict__ cols,
                             const float* __restrict__ dinv, float* __restrict__ tx,
                             int ld, int srcOff, int dstOff, int E, int F) {
  int i = blockIdx.x * blockDim.x + threadIdx.x;
  if (i < E * F) {
    int e = i / F, f = i - e * F;
    int r = rows[e], c = cols[e];
    float w = -dinv[r] * dinv[c];
    atomicAdd(&tx[r * ld + dstOff + f], w * tx[c * ld + srcOff + f]);
  }
}

// Chebyshev recurrence finalize: T_k = 2*Lhat(T_{k-1}) - T_{k-2}
__global__ void cheb_update(float* __restrict__ tx, int ld, int kOff, int k2Off, int n, int F) {
  int i = blockIdx.x * blockDim.x + threadIdx.x;
  if (i < n * F) {
    int r = i / F, f = i - r * F;
    float* p = tx + r * ld;
    p[kOff + f] = 2.0f * p[kOff + f] - p[k2Off + f];
  }
}

// dst[rows[i]*F + f] += vals[i] * src[cols[i]*F + f]
__global__ void pool_scatter(const int* __restrict__ rows, const int* __restrict__ cols,
                             const float* __restrict__ vals, const float* __restrict__ src,
                             float* __restrict__ dst, int n, int F) {
  int i = blockIdx.x * blockDim.x + threadIdx.x;
  if (i < n * F) {
    int e = i / F, f = i - e * F;
    atomicAdd(&dst[rows[e] * F + f], vals[e] * src[cols[e] * F + f]);
  }
}

// Pack B[K,N] (zero-padded to Kpad rows) into fragment order:
//   Bp[(k>>1)*2N + 2n + (k&1)] = B[k,n]
// so a lane's B pair (rows kk,kk+1 at col n) is one aligned b64 load.
__global__ void pack_b(const float* __restrict__ B, float* __restrict__ Bp,
                       int Kreal, int Kpad, int N) {
  int i = blockIdx.x * blockDim.x + threadIdx.x;
  if (i < Kpad * N) {
    int k = i / N, n = i - k * N;
    float v = (k < Kreal) ? B[k * N + n] : 0.0f;
    Bp[(k >> 1) * (2 * N) + 2 * n + (k & 1)] = v;
  }
}

// ---------------------------------------------------------------------------
// Dense GEMM via V_WMMA_F32_16X16X4_F32 (one 16x16 C tile per wave32)
//   A[M,Kpad] row-major (rows padded to mult-of-16, readable garbage ok),
//   Bp packed as above (pad K rows are zero), C[M,N] = A@B + bias (+ReLU).
// Hot loop: one b64 A load + one b64 B load + wmma, no branches.
// Fragment layouts per CDNA5 ISA 7.12.2.
// ---------------------------------------------------------------------------

template <bool RELU>
__global__ __launch_bounds__(256) void wmma_gemm_bias(
    const float* __restrict__ A, int lda,
    const float* __restrict__ Bp,
    const float* __restrict__ bias, float* __restrict__ C,
    int M, int Kpad, int N) {
  int wave = (blockIdx.x * blockDim.x + threadIdx.x) >> 5;
  int lane = threadIdx.x & 31;
  int tilesN = N >> 4;
  int tilesM = (M + 15) >> 4;
  if (wave >= tilesM * tilesN) return;  // uniform per wave -> EXEC all-ones at WMMA
  int tm = wave / tilesN, tn = wave - tm * tilesN;
  int m0 = tm << 4, n0 = tn << 4;
  int half = lane >> 4;  // selects K pair (A/B) and row half (C)
  int l16 = lane & 15;

  const float* aPtr = A + (long)(m0 + l16) * lda + 2 * half;  // + k0
  const float* bPtr = Bp + (long)half * 2 * N + 2 * (n0 + l16);  // + (k0/2)*2N

  v8f c = {};
#pragma unroll 4
  for (int k0 = 0; k0 < Kpad; k0 += 4) {
    __builtin_prefetch(aPtr + k0 + 64, 0, 0);
    v2f a = *(const v2f*)(aPtr + k0);
    v2f b = *(const v2f*)(bPtr + (k0 >> 1) * (2 * N));
    c = __builtin_amdgcn_wmma_f32_16x16x4_f32(false, a, false, b, (short)0, c, false, false);
  }

  int col = n0 + l16;
  float bv = bias[col];
#pragma unroll
  for (int i = 0; i < 8; ++i) {
    int row = m0 + i + half * 8;
    if (row < M) {
      float v = c[i] + bv;
      if (RELU) v = fmaxf(v, 0.0f);
      C[(long)row * N + col] = v;
    }
  }
}

// ---------------------------------------------------------------------------
// Head: [M,128] @ [128,1024] + bh, fused global column-max into gmax[1024]
// ---------------------------------------------------------------------------

__device__ __forceinline__ void atomicMaxF(float* addr, float val) {
  // sign-aware float atomic max (positives: int max; negatives: uint min)
  if (val >= 0.0f)
    atomicMax((int*)addr, __float_as_int(val));
  else
    atomicMin((unsigned int*)addr, (unsigned int)__float_as_int(val));
}

__global__ __launch_bounds__(256) void wmma_head_max(
    const float* __restrict__ A, const float* __restrict__ Bp,
    const float* __restrict__ bias, float* __restrict__ gmax, int M) {
  const int K = 128, N = 1024;
  int wave = (blockIdx.x * blockDim.x + threadIdx.x) >> 5;
  int lane = threadIdx.x & 31;
  int tilesN = N >> 4;
  int tilesM = (M + 15) >> 4;
  if (wave >= tilesM * tilesN) return;
  int tm = wave / tilesN, tn = wave - tm * tilesN;
  int m0 = tm << 4, n0 = tn << 4;
  int half = lane >> 4;
  int l16 = lane & 15;

  const float* aPtr = A + (long)(m0 + l16) * K + 2 * half;
  const float* bPtr = Bp + (long)half * 2 * N + 2 * (n0 + l16);

  v8f c = {};
#pragma unroll 4
  for (int k0 = 0; k0 < K; k0 += 4) {
    v2f a = *(const v2f*)(aPtr + k0);
    v2f b = *(const v2f*)(bPtr + (k0 >> 1) * (2 * N));
    c = __builtin_amdgcn_wmma_f32_16x16x4_f32(false, a, false, b, (short)0, c, false, false);
  }

  int col = n0 + l16;
  float bv = bias[col];
  float m = -INFINITY;
#pragma unroll
  for (int i = 0; i < 8; ++i) {
    int row = m0 + i + half * 8;
    if (row < M) m = fmaxf(m, c[i] + bv);  // mask pad rows out of the max
  }
  // lanes l and l+16 hold the same output column -> combine halves
  m = fmaxf(m, __shfl_xor(m, 16, 32));
  if (half == 0) atomicMaxF(&gmax[col], m);
}

// out[c] = bl[c] + sum_n gmax[n] * Wl[n, c]
__global__ void final_fc(const float* __restrict__ gmax, const float* __restrict__ Wl,
                         const float* __restrict__ bl, float* __restrict__ out) {
  int c = blockIdx.x * blockDim.x + threadIdx.x;
  if (c < NCLS) {
    float acc = bl[c];
    for (int n = 0; n < 1024; ++n) acc += gmax[n] * Wl[n * NCLS + c];
    out[c] = acc;
  }
}

// ---------------------------------------------------------------------------
// Orchestration
// ---------------------------------------------------------------------------

extern "C" void kernel_launch(void* const* d_in, const int* in_sizes, int n_in,
                              void* d_out, int out_size, void* d_ws, size_t ws_size,
                              hipStream_t stream) {
  (void)in_sizes; (void)n_in; (void)out_size; (void)ws_size;

  const float* pos     = (const float*)d_in[0];
  const int*   ei0     = (const int*)d_in[1];
  const int*   ei1     = (const int*)d_in[2];
  const int*   ei2     = (const int*)d_in[3];
  const int*   ds0_row = (const int*)d_in[4];
  const int*   ds0_col = (const int*)d_in[5];
  const float* ds0_val = (const float*)d_in[6];
  const int*   ds1_row = (const int*)d_in[7];
  const int*   ds1_col = (const int*)d_in[8];
  const float* ds1_val = (const float*)d_in[9];
  const float* W0 = (const float*)d_in[10];
  const float* b0 = (const float*)d_in[11];
  const float* W1 = (const float*)d_in[12];
  const float* b1 = (const float*)d_in[13];
  const float* W2 = (const float*)d_in[14];
  const float* b2 = (const float*)d_in[15];
  const float* Wh = (const float*)d_in[16];
  const float* bh = (const float*)d_in[17];
  const float* Wl = (const float*)d_in[18];
  const float* bl = (const float*)d_in[19];
  float* out = (float*)d_out;

  // Workspace carve-up (floats); regions reused across stages.
  float* ws    = (float*)d_ws;
  float* dinv  = ws;                  //   100,000
  float* txcat = dinv + 100000;       // 4,801,536 (max: 25008 rows * 192)
  float* hbuf  = txcat + 4801536;     // 3,200,000 (max: N0*32; head pad reads fit)
  float* pbuf  = hbuf + 3200000;      //   800,000 (max: N1*32)
  float* wpack = pbuf + 800000;       //   131,072 (max: head 128*1024)
  float* gmax  = wpack + 131072;      //     1,024

  auto cdiv = [](int a, int b) { return (a + b - 1) / b; };
  const int T = 256;

  // One ChebConv stage: out[n,Cout] = relu?( concat_k(T_k) @ W[K*Cin,Cout] + b )
  auto run_cheb = [&](const float* x, int n, int Cin, int Cout,
                      const int* ei, int E, const float* W, const float* b,
                      float* outp, bool relu) {
    const int* rows = ei;
    const int* cols = ei + E;
    int kReal = KCH * Cin;
    int kPad  = (kReal + 3) & ~3;       // lda, multiple of 4
    int nPad  = (n + 15) & ~15;         // rows padded for unconditional A loads

    fill_f32<<<cdiv(n, T), T, 0, stream>>>(dinv, 0.0f, n);
    deg_accum<<<cdiv(E, T), T, 0, stream>>>(rows, dinv, E);
    deg_to_dinv<<<cdiv(n, T), T, 0, stream>>>(dinv, n);

    fill_f32<<<cdiv(nPad * kPad, T), T, 0, stream>>>(txcat, 0.0f, nPad * kPad);
    copy_rows<<<cdiv(n * Cin, T), T, 0, stream>>>(x, Cin, txcat, kPad, n, Cin);  // T_0 = x

    for (int k = 1; k < KCH; ++k) {
      lhat_scatter<<<cdiv(E * Cin, T), T, 0, stream>>>(
          rows, cols, dinv, txcat, kPad, (k - 1) * Cin, k * Cin, E, Cin);
      if (k >= 2)
        cheb_update<<<cdiv(n * Cin, T), T, 0, stream>>>(
            txcat, kPad, k * Cin, (k - 2) * Cin, n, Cin);
    }

    // Pack weights [kReal,Cout] -> fragment-friendly, zero pad rows to kPad.
    pack_b<<<cdiv(kPad * Cout, T), T, 0, stream>>>(W, wpack, kReal, kPad, Cout);

    int tiles = (nPad >> 4) * (Cout >> 4);
    int blocks = cdiv(tiles * 32, T);
    if (relu)
      wmma_gemm_bias<true><<<blocks, T, 0, stream>>>(txcat, kPad, wpack, b, outp, n, kPad, Cout);
    else
      wmma_gemm_bias<false><<<blocks, T, 0, stream>>>(txcat, kPad, wpack, b, outp, n, kPad, Cout);
  };

  // Stage 0: pos[N0,3] -> h[N0,32], ReLU
  run_cheb(pos, CN0, 3, 32, ei0, CE0, W0, b0, hbuf, true);

  // Pool 0 -> pbuf[N1,32]
  fill_f32<<<cdiv(CN1 * 32, T), T, 0, stream>>>(pbuf, 0.0f, CN1 * 32);
  pool_scatter<<<cdiv(CN0 * 32, T), T, 0, stream>>>(ds0_row, ds0_col, ds0_val, hbuf, pbuf, CN0, 32);

  // Stage 1: [N1,32] -> h[N1,64], ReLU
  run_cheb(pbuf, CN1, 32, 64, ei1, CE1, W1, b1, hbuf, true);

  // Pool 1 -> pbuf[N2,64]
  fill_f32<<<cdiv(CN2 * 64, T), T, 0, stream>>>(pbuf, 0.0f, CN2 * 64);
  pool_scatter<<<cdiv(CN1 * 64, T), T, 0, stream>>>(ds1_row, ds1_col, ds1_val, hbuf, pbuf, CN1, 64);

  // Stage 2: [N2,64] -> h[N2,128], no ReLU
  run_cheb(pbuf, CN2, 64, 128, ei2, CE2, W2, b2, hbuf, false);

  // Head: [N2,128]@[128,1024]+bh, fused global max over nodes
  pack_b<<<cdiv(128 * 1024, T), T, 0, stream>>>(Wh, wpack, 128, 128, 1024);
  fill_f32<<<cdiv(1024, T), T, 0, stream>>>(gmax, -INFINITY, 1024);
  {
    int tiles = cdiv(CN2, 16) * (1024 / 16);
    wmma_head_max<<<cdiv(tiles * 32, T), T, 0, stream>>>(hbuf, wpack, bh, gmax, CN2);
  }

  // Classifier: gmax[1024]@Wl[1024,40]+bl -> out[40]
  final_fc<<<1, 64, 0, stream>>>(gmax, Wl, bl, out);
}